// MessagePassingEncoder_10539849744616
// MI455X (gfx1250) — compile-verified
//
#include <hip/hip_runtime.h>

#define NN 50000
#define NE 800000
#define DIM 128

typedef float v2f __attribute__((ext_vector_type(2)));
typedef float v8f __attribute__((ext_vector_type(8)));

__global__ void zero_kernel(float* __restrict__ p, int n) {
    int i = blockIdx.x * blockDim.x + threadIdx.x;
    if (i < n) p[i] = 0.0f;
}

__global__ void deg_kernel(const long long* __restrict__ dst,
                           float* __restrict__ deg, int nE) {
    int i = blockIdx.x * blockDim.x + threadIdx.x;
    if (i < nE) unsafeAtomicAdd(&deg[dst[i]], 1.0f);
}

__global__ void invdeg_kernel(float* __restrict__ deg, int n) {
    int i = blockIdx.x * blockDim.x + threadIdx.x;
    if (i < n) deg[i] = 1.0f / fmaxf(deg[i], 1.0f);
}

// 32 lanes per edge, float4 per lane: gather h[src], atomic-add into agg[dst].
// agg is 25.6MB -> L2 resident; f32 hardware atomics resolve at L2.
__global__ void scatter_kernel(const float* __restrict__ h,
                               const long long* __restrict__ src,
                               const long long* __restrict__ dst,
                               float* __restrict__ agg, int nE) {
    long long i = (long long)blockIdx.x * blockDim.x + threadIdx.x;
    const long long total  = (long long)nE * 32;
    const long long stride = (long long)gridDim.x * blockDim.x;
    for (; i < total; i += stride) {
        const int e    = (int)(i >> 5);
        const int lane = (int)(i & 31);
        const long long s = src[e];
        const long long d = dst[e];
        const float4 v = *reinterpret_cast<const float4*>(h + s * DIM + lane * 4);
        float* p = agg + d * DIM + lane * 4;
        unsafeAtomicAdd(p + 0, v.x);
        unsafeAtomicAdd(p + 1, v.y);
        unsafeAtomicAdd(p + 2, v.z);
        unsafeAtomicAdd(p + 3, v.w);
    }
}

// One block = 16 output rows x 128 cols. 8 waves, wave w owns cols [16w,16w+16).
// out = (agg*inv_deg) @ Wl^T + bias + hin @ Wr^T, optional ReLU.
// Uses V_WMMA_F32_16X16X4_F32 (exact fp32 accumulate).
__global__ __launch_bounds__(256) void sage_gemm_kernel(
    const float* __restrict__ agg, const float* __restrict__ inv_deg,
    const float* __restrict__ hin,
    const float* __restrict__ Wl, const float* __restrict__ bl,
    const float* __restrict__ Wr,
    float* __restrict__ hout, int relu)
{
    __shared__ float sA[16 * 129];   // agg * inv_deg, padded rows (bank spread)
    __shared__ float sH[16 * 129];   // hin panel

    const int row0 = blockIdx.x * 16;
    const int t = threadIdx.x;

    // Stage both 16x128 panels (coalesced 128-wide loads).
    for (int idx = t; idx < 16 * 128; idx += 256) {
        const int r = idx >> 7;
        const int c = idx & 127;
        const int g = row0 + r;
        const float s = inv_deg[g];
        sA[r * 129 + c] = agg[(long long)g * DIM + c] * s;
        sH[r * 129 + c] = hin[(long long)g * DIM + c];
    }
    __syncthreads();

    const int lane = t & 31;
    const int wave = t >> 5;
    const int n    = lane & 15;   // N index (and B/W row)
    const int half = lane >> 4;   // K sub-select for A/B frags
    const int j0   = wave * 16;   // output column tile

    v8f acc = {};

    // D += A(16x4) x Wl^T(4x16), K = 0..127
    for (int k0 = 0; k0 < DIM; k0 += 4) {
        const int k = k0 + 2 * half;
        v2f a, b;
        a[0] = sA[n * 129 + k];
        a[1] = sA[n * 129 + k + 1];
        b[0] = Wl[(j0 + n) * DIM + k];      // B[k][j] = Wl[j][k]
        b[1] = Wl[(j0 + n) * DIM + k + 1];
        acc = __builtin_amdgcn_wmma_f32_16x16x4_f32(false, a, false, b,
                                                    (short)0, acc, false, false);
    }
    // D += H(16x4) x Wr^T(4x16)
    for (int k0 = 0; k0 < DIM; k0 += 4) {
        const int k = k0 + 2 * half;
        v2f a, b;
        a[0] = sH[n * 129 + k];
        a[1] = sH[n * 129 + k + 1];
        b[0] = Wr[(j0 + n) * DIM + k];
        b[1] = Wr[(j0 + n) * DIM + k + 1];
        acc = __builtin_amdgcn_wmma_f32_16x16x4_f32(false, a, false, b,
                                                    (short)0, acc, false, false);
    }

    // C/D layout: VGPR r, lane(n,half) -> element (r + 8*half, n)
    const float bias = bl[j0 + n];
    for (int r = 0; r < 8; ++r) {
        float v = acc[r] + bias;
        if (relu) v = fmaxf(v, 0.0f);
        hout[(long long)(row0 + r + 8 * half) * DIM + j0 + n] = v;
    }
}

extern "C" void kernel_launch(void* const* d_in, const int* in_sizes, int n_in,
                              void* d_out, int out_size, void* d_ws, size_t ws_size,
                              hipStream_t stream) {
    const float*     x   = (const float*)d_in[0];
    const long long* ei  = (const long long*)d_in[1];   // int64 [2][E]
    const float*     W_l = (const float*)d_in[2];       // [3][128][128]
    const float*     b_l = (const float*)d_in[3];       // [3][128]
    const float*     W_r = (const float*)d_in[4];       // [3][128][128]
    float*           out = (float*)d_out;

    const long long* src = ei;
    const long long* dst = ei + NE;

    // Workspace: inv_deg | agg | h ping buffer  (~51.4 MB)
    float* deg  = (float*)d_ws;                       // NN floats -> inv_deg
    float* agg  = deg + 50176;                        // 256-aligned
    float* hbuf = agg + (long long)NN * DIM;

    // Degrees (invariant across layers).
    zero_kernel<<<(NN + 255) / 256, 256, 0, stream>>>(deg, NN);
    deg_kernel<<<(NE + 255) / 256, 256, 0, stream>>>(dst, deg, NE);
    invdeg_kernel<<<(NN + 255) / 256, 256, 0, stream>>>(deg, NN);

    const float* hin = x;
    for (int layer = 0; layer < 3; ++layer) {
        float* hout = (layer == 1) ? hbuf : out;      // x -> out -> hbuf -> out
        zero_kernel<<<(NN * DIM + 255) / 256, 256, 0, stream>>>(agg, NN * DIM);
        scatter_kernel<<<4096, 256, 0, stream>>>(hin, src, dst, agg, NE);
        sage_gemm_kernel<<<NN / 16, 256, 0, stream>>>(
            agg, deg, hin,
            W_l + (long long)layer * DIM * DIM,
            b_l + layer * DIM,
            W_r + (long long)layer * DIM * DIM,
            hout, (layer < 2) ? 1 : 0);
        hin = hout;
    }
}